// GruFlowtron_64707977281849
// MI455X (gfx1250) — compile-verified
//
#include <hip/hip_runtime.h>
#include <math.h>

#define NUM_LAYERS 8
#define HIDDEN     1000
#define GATES      (3 * HIDDEN)   /* 3000 rows */
#define K_DIM      HIDDEN         /* 1000 */
#define KCHUNKS    5
#define KCHUNK     200            /* 1000/5, multiple of 8 -> keeps 8B alignment */

typedef float v2f __attribute__((ext_vector_type(2)));
typedef float v8f __attribute__((ext_vector_type(8)));

// ---------------------------------------------------------------------------
// Partial GEMV: y_part[z][row] = W[layer][row, z*KCHUNK .. +KCHUNK) @ x[...]
// 256 threads = 8 waves; each wave computes a 16-row tile over one K-chunk
// with V_WMMA_F32_16X16X4_F32 (A = 16x4 weight tile per ISA layout, B = x
// chunk broadcast across columns). grid = (24 row-blocks, layers, KCHUNKS).
// Clamped loads / guarded stores keep EXEC all-ones through every WMMA.
// ---------------------------------------------------------------------------
__global__ __launch_bounds__(256)
void gru_gemv3k_wmma(const float* __restrict__ Wbase,
                     const float* __restrict__ xbase,
                     float*       __restrict__ ybase,   // [layer][part][GATES]
                     long wStride, long xStride)
{
    __shared__ float xs[KCHUNK];

    const int layer  = blockIdx.y;
    const int kc     = blockIdx.z;
    const int kStart = kc * KCHUNK;

    const float* W = Wbase + (long)layer * wStride;
    const float* x = xbase + (long)layer * xStride + kStart;
    float*       y = ybase + ((long)layer * KCHUNKS + kc) * GATES;

    for (int j = threadIdx.x; j < KCHUNK; j += blockDim.x) xs[j] = x[j];
    __syncthreads();

    const int wave = threadIdx.x >> 5;       // 0..7
    const int lid  = threadIdx.x & 31;
    const int g    = lid >> 4;               // half-wave 0/1 (ISA A layout: K pairs)
    const int m    = lid & 15;               // row-in-tile

    const int rowBase    = (blockIdx.x * 8 + wave) * 16;
    const int row        = rowBase + m;
    const int rowClamped = (row < GATES) ? row : (GATES - 1);
    const float* wrow    = W + (long)rowClamped * K_DIM + kStart + 2 * g;

    v8f acc0 = {0.f, 0.f, 0.f, 0.f, 0.f, 0.f, 0.f, 0.f};
    v8f acc1 = {0.f, 0.f, 0.f, 0.f, 0.f, 0.f, 0.f, 0.f};

    for (int k0 = 0; k0 < KCHUNK; k0 += 8) {
        // keep the weight stream ahead in GL2 -> global_prefetch_b8
        __builtin_prefetch(wrow + k0 + 64, 0, 0);

        // A tiles: lane 16g+m holds W[m, k+2g], W[m, k+2g+1]  (8B loads)
        v2f a0 = *(const v2f*)(wrow + k0);
        v2f a1 = *(const v2f*)(wrow + k0 + 4);
        // B tiles: B[k][n] = x[k] for all n (broadcast columns) -> ds_load_b64
        v2f b0 = *(const v2f*)(xs + k0 + 2 * g);
        v2f b1 = *(const v2f*)(xs + k0 + 4 + 2 * g);

        acc0 = __builtin_amdgcn_wmma_f32_16x16x4_f32(false, a0, false, b0,
                                                     (short)0, acc0, false, false);
        acc1 = __builtin_amdgcn_wmma_f32_16x16x4_f32(false, a1, false, b1,
                                                     (short)0, acc1, false, false);
    }

    // D layout: VGPR r, lanes 0-15 = (M=r, N=lane); lanes 16-31 = (M=r+8).
    // All N columns identical; lanes with m==0 hold the result column.
    if (m == 0) {
        #pragma unroll
        for (int r = 0; r < 8; ++r) {
            int outRow = rowBase + 8 * g + r;
            if (outRow < GATES)
                y[outRow] = acc0[r] + acc1[r];
        }
    }
}

// ---------------------------------------------------------------------------
// Gate math: sums the KCHUNKS partials, adds biases, applies GRU gates.
// Layer 0's gi is a 3000x2 mat-vec -> computed inline from W_ih0.
// ---------------------------------------------------------------------------
__global__ __launch_bounds__(256)
void gru_combine(const float* __restrict__ gi_parts,  // [KCHUNKS][GATES]
                 const float* __restrict__ gh_parts,  // [KCHUNKS][GATES]
                 const float* __restrict__ bih,       // [GATES] for this layer
                 const float* __restrict__ bhh,       // [GATES] for this layer
                 const float* __restrict__ hprev,     // [HIDDEN]
                 float*       __restrict__ hout,      // [HIDDEN]
                 const float* __restrict__ Wih0,      // (GATES,2) or null
                 const float* __restrict__ x0,        // (2,) or null
                 int isLayer0)
{
    for (int j = threadIdx.x + blockIdx.x * blockDim.x; j < HIDDEN;
         j += blockDim.x * gridDim.x) {
        float gir, giz, gin;
        if (isLayer0) {
            float a = x0[0], c = x0[1];
            gir = Wih0[(long)(j             ) * 2] * a + Wih0[(long)(j             ) * 2 + 1] * c;
            giz = Wih0[(long)(j +     HIDDEN) * 2] * a + Wih0[(long)(j +     HIDDEN) * 2 + 1] * c;
            gin = Wih0[(long)(j + 2 * HIDDEN) * 2] * a + Wih0[(long)(j + 2 * HIDDEN) * 2 + 1] * c;
        } else {
            gir = 0.f; giz = 0.f; gin = 0.f;
            #pragma unroll
            for (int p = 0; p < KCHUNKS; ++p) {
                gir += gi_parts[(long)p * GATES + j];
                giz += gi_parts[(long)p * GATES + j + HIDDEN];
                gin += gi_parts[(long)p * GATES + j + 2 * HIDDEN];
            }
        }
        gir += bih[j];
        giz += bih[j + HIDDEN];
        gin += bih[j + 2 * HIDDEN];

        float ghr = bhh[j], ghz = bhh[j + HIDDEN], ghn = bhh[j + 2 * HIDDEN];
        #pragma unroll
        for (int p = 0; p < KCHUNKS; ++p) {
            ghr += gh_parts[(long)p * GATES + j];
            ghz += gh_parts[(long)p * GATES + j + HIDDEN];
            ghn += gh_parts[(long)p * GATES + j + 2 * HIDDEN];
        }

        float r = 1.f / (1.f + __expf(-(gir + ghr)));
        float z = 1.f / (1.f + __expf(-(giz + ghz)));
        float n = tanhf(gin + r * ghn);
        hout[j] = (1.f - z) * n + z * hprev[j];
    }
}

// ---------------------------------------------------------------------------
// flow = W_out @ h7 + b_out
// ---------------------------------------------------------------------------
__global__ __launch_bounds__(256)
void gru_outdot(const float* __restrict__ Wout,
                const float* __restrict__ h,
                const float* __restrict__ bout,
                float*       __restrict__ out)
{
    __shared__ float red[8];
    float s = 0.f;
    for (int j = threadIdx.x; j < HIDDEN; j += 256) s += Wout[j] * h[j];
    #pragma unroll
    for (int off = 16; off > 0; off >>= 1) s += __shfl_down(s, off, 32);
    if ((threadIdx.x & 31) == 0) red[threadIdx.x >> 5] = s;
    __syncthreads();
    if (threadIdx.x == 0) {
        float t = 0.f;
        #pragma unroll
        for (int w = 0; w < 8; ++w) t += red[w];
        out[0] = t + bout[0];
    }
}

// ---------------------------------------------------------------------------
extern "C" void kernel_launch(void* const* d_in, const int* in_sizes, int n_in,
                              void* d_out, int out_size, void* d_ws, size_t ws_size,
                              hipStream_t stream)
{
    (void)in_sizes; (void)n_in; (void)out_size; (void)ws_size;

    const float* input   = (const float*)d_in[0];  // (2,)
    const float* hiddens = (const float*)d_in[1];  // (8,1,1000)
    const float* W_ih0   = (const float*)d_in[2];  // (3000,2)
    const float* W_ih    = (const float*)d_in[3];  // (7,3000,1000)
    const float* W_hh    = (const float*)d_in[4];  // (8,3000,1000)
    const float* b_ih    = (const float*)d_in[5];  // (8,3000)
    const float* b_hh    = (const float*)d_in[6];  // (8,3000)
    const float* W_out   = (const float*)d_in[7];  // (1,1000)
    const float* b_out   = (const float*)d_in[8];  // (1,)

    float* out = (float*)d_out;                          // [0]=flow, [1..8000]=h'
    float* gh  = (float*)d_ws;                           // 8*5*3000 floats
    float* gi  = gh + (long)NUM_LAYERS * KCHUNKS * GATES;// 8*5*3000 floats

    const dim3 blk(256);

    // 1) All W_hh mat-vecs are independent of the layer chain -> one wide
    //    launch: 24 row-blocks x 8 layers x 5 K-chunks = 960 blocks.
    gru_gemv3k_wmma<<<dim3(24, NUM_LAYERS, KCHUNKS), blk, 0, stream>>>(
        W_hh, hiddens, gh, (long)GATES * K_DIM, (long)HIDDEN);

    // 2) Layer 0 (input dim 2): gates directly.
    gru_combine<<<dim3(4), blk, 0, stream>>>(
        nullptr, gh /* layer 0 parts */, b_ih, b_hh,
        hiddens, out + 1, W_ih0, input, 1);

    // 3) Sequential W_ih chain; stream order provides inter-layer sync.
    //    K-split to 5 partials -> 120 blocks (960 waves) per dependent GEMV.
    for (int l = 1; l < NUM_LAYERS; ++l) {
        gru_gemv3k_wmma<<<dim3(24, 1, KCHUNKS), blk, 0, stream>>>(
            W_ih + (long)(l - 1) * GATES * K_DIM,
            out + 1 + (long)(l - 1) * HIDDEN,
            gi + (long)l * KCHUNKS * GATES,
            0L, 0L);
        gru_combine<<<dim3(4), blk, 0, stream>>>(
            gi + (long)l * KCHUNKS * GATES,
            gh + (long)l * KCHUNKS * GATES,
            b_ih + (long)l * GATES, b_hh + (long)l * GATES,
            hiddens + (long)l * HIDDEN, out + 1 + (long)l * HIDDEN,
            nullptr, nullptr, 0);
    }

    // 4) flow = W_out @ h7 + b_out
    gru_outdot<<<dim3(1), blk, 0, stream>>>(
        W_out, out + 1 + (long)(NUM_LAYERS - 1) * HIDDEN, b_out, out);
}